// GraphProposalNetwork_36825049596052
// MI455X (gfx1250) — compile-verified
//
#include <hip/hip_runtime.h>

typedef __attribute__((ext_vector_type(2))) float v2f;
typedef __attribute__((ext_vector_type(8))) float v8f;

// D = A(16x4) * B(4x16) + C, fp32 WMMA (wave32)
#define WMMA_F32(a, b, c) \
  __builtin_amdgcn_wmma_f32_16x16x4_f32(false, (a), false, (b), (short)0, (c), false, false)

// ---------------------------------------------------------------------------
// Kernel 1: E = X @ Wp + bp
//   X : (10240, 2048) f32, Wp: (2048,128), bp: (128) -> E: (10240,128)
// Block = 64 rows x 128 cols, 256 threads (8 waves).
// Each wave owns one 16-col N-tile and FOUR 16-row M-tiles:
//   -> 4 independent WMMA accumulation chains per wave (ILP for the XDL pipe)
//   -> each B-operand load is reused 4x (Wp L2 traffic: 160 MB total)
// K staged through a 64x64 LDS chunk; next chunk prefetched during compute.
// ---------------------------------------------------------------------------
__global__ __launch_bounds__(256) void k_emb(const float* __restrict__ X,
                                             const float* __restrict__ Wp,
                                             const float* __restrict__ bp,
                                             float* __restrict__ E) {
  __shared__ float Asm[64 * 64];            // 64 rows x 64-wide K chunk (16 KB)
  const int tid    = threadIdx.x;
  const int lane   = tid & 31;
  const int wave   = tid >> 5;              // 0..7 -> N tile
  const int m_base = blockIdx.x * 64;
  const int col    = wave * 16 + (lane & 15);
  const int koff   = (lane >> 4) * 2;       // 0 or 2 (lane-half selects K pair)
  const int mrow   = lane & 15;

  v8f acc0 = {}, acc1 = {}, acc2 = {}, acc3 = {};
  for (int kc = 0; kc < 2048; kc += 64) {
    // stage 64x64 A chunk: 1024 float4s, 4 per thread, fully coalesced
#pragma unroll
    for (int q = 0; q < 4; ++q) {
      const int linear = q * 256 + tid;
      const int r  = linear >> 4;           // 0..63
      const int kk = (linear & 15) * 4;     // 0..60
      const float4 v =
          *(const float4*)(X + (size_t)(m_base + r) * 2048 + kc + kk);
      *(float4*)(Asm + r * 64 + kk) = v;
      // prefetch next K-chunk of X (global_prefetch_b8)
      if (kc + 64 < 2048)
        __builtin_prefetch(X + (size_t)(m_base + r) * 2048 + kc + 64 + kk, 0, 1);
    }
    __syncthreads();
#pragma unroll
    for (int ks = 0; ks < 64; ks += 4) {
      v2f b;
      const float* wp = Wp + (size_t)(kc + ks + koff) * 128 + col;
      b.x = wp[0];
      b.y = wp[128];
      v2f a0, a1, a2, a3;
      const int base = mrow * 64 + ks + koff;
      a0.x = Asm[base];             a0.y = Asm[base + 1];
      a1.x = Asm[base + 16 * 64];   a1.y = Asm[base + 16 * 64 + 1];
      a2.x = Asm[base + 32 * 64];   a2.y = Asm[base + 32 * 64 + 1];
      a3.x = Asm[base + 48 * 64];   a3.y = Asm[base + 48 * 64 + 1];
      acc0 = WMMA_F32(a0, b, acc0);
      acc1 = WMMA_F32(a1, b, acc1);
      acc2 = WMMA_F32(a2, b, acc2);
      acc3 = WMMA_F32(a3, b, acc3);
    }
    __syncthreads();
  }
  // C/D layout: VGPR r, lane -> M = r + 8*(lane>=16), N = col
  const float bias = bp[col];
  const int mh = (lane >> 4) * 8;
#pragma unroll
  for (int r = 0; r < 8; ++r) {
    const size_t row = (size_t)(m_base + mh + r);
    E[(row          ) * 128 + col] = acc0[r] + bias;
    E[(row + 16     ) * 128 + col] = acc1[r] + bias;
    E[(row + 32     ) * 128 + col] = acc2[r] + bias;
    E[(row + 48     ) * 128 + col] = acc3[r] + bias;
  }
}

// ---------------------------------------------------------------------------
// Kernel 2 (fused): block handles TWO batches x one head:
//   A  = emb[b] @ W1[h, :128, :] + b1[h]   (10x128 per batch, padded to 16)
//   Bm = emb[b] @ W1[h, 128:, :]
//   S[i,k] = sum_f lrelu(A[i,f] + Bm[k,f]) * W2[h,f] + b2[h]
//   adj[b,h,j,i] = S[i, j + (j>=i)] for j<9 ; row j=9 zeroed
// 2 batches/block halves W1 L2 traffic (1 GB -> 512 MB) and gives each wave
// 4 independent WMMA chains (2 M-tiles x {top,bottom}).
// Grid: (b-group, h) with b fastest -> consecutive blocks reuse W1[h] in L2.
// ---------------------------------------------------------------------------
__global__ __launch_bounds__(256) void k_attn(const float* __restrict__ E,   // (1024,10,128)
                                              const float* __restrict__ W1,  // (8,256,128)
                                              const float* __restrict__ b1,  // (8,128)
                                              const float* __restrict__ W2,  // (8,128)
                                              const float* __restrict__ b2,  // (8)
                                              float* __restrict__ out) {     // (1024,8,10,10)
  __shared__ float Es[32 * 128];  // 2 batches, each padded to 16 rows (16 KB)
  __shared__ float As[2 * 10 * 128];
  __shared__ float Bs[2 * 10 * 128];
  __shared__ float W2s[128];
  __shared__ float Ss[200];

  const int bg   = blockIdx.x;       // batch group (2 batches)
  const int h    = blockIdx.y;
  const int b0   = bg * 2;
  const int tid  = threadIdx.x;
  const int lane = tid & 31;
  const int wave = tid >> 5;

  // load emb for both batches; rows 10..15 of each 16-row slot zero-padded
  for (int idx = tid; idx < 32 * 128; idx += 256) {
    const int slot = idx >> 11;              // 0 or 1 (batch)
    const int rem  = idx & 2047;             // row*128+f within slot
    const int row  = rem >> 7;
    Es[idx] = (row < 10) ? E[(size_t)(b0 + slot) * 1280 + rem] : 0.0f;
  }
  if (tid < 128) W2s[tid] = W2[h * 128 + tid];
  __syncthreads();

  const int col  = wave * 16 + (lane & 15);
  const int koff = (lane >> 4) * 2;
  const int mrow = lane & 15;
  const float* w1h = W1 + (size_t)h * 256 * 128;

  v8f cA0 = {}, cB0 = {}, cA1 = {}, cB1 = {};
#pragma unroll 4
  for (int k0 = 0; k0 < 128; k0 += 4) {
    v2f bT, bB;
    const float* wt = w1h + (size_t)(k0 + koff) * 128 + col; // top half
    bT.x = wt[0];
    bT.y = wt[128];
    const float* wb = wt + 128 * 128;                        // bottom half
    bB.x = wb[0];
    bB.y = wb[128];
    v2f a0, a1;
    const int base = mrow * 128 + k0 + koff;
    a0.x = Es[base];             a0.y = Es[base + 1];
    a1.x = Es[base + 16 * 128];  a1.y = Es[base + 16 * 128 + 1];
    cA0 = WMMA_F32(a0, bT, cA0);
    cB0 = WMMA_F32(a0, bB, cB0);
    cA1 = WMMA_F32(a1, bT, cA1);
    cB1 = WMMA_F32(a1, bB, cB1);
  }

  const float b1v = b1[h * 128 + col];
  const int mh = (lane >> 4) * 8;
#pragma unroll
  for (int r = 0; r < 8; ++r) {
    const int m = mh + r;
    if (m < 10) {
      As[(0 * 10 + m) * 128 + col] = cA0[r] + b1v;  // fold b1 into A side
      Bs[(0 * 10 + m) * 128 + col] = cB0[r];
      As[(1 * 10 + m) * 128 + col] = cA1[r] + b1v;
      Bs[(1 * 10 + m) * 128 + col] = cB1[r];
    }
  }
  __syncthreads();

  // 200 pairs (2 batches x 100), one per thread; staggered LDS-bank walk
  if (tid < 200) {
    const int bb  = tid / 100;
    const int rem = tid % 100;
    const int i   = rem / 10;
    const int k   = rem % 10;
    const float* ai = As + (bb * 10 + i) * 128;
    const float* bk = Bs + (bb * 10 + k) * 128;
    float acc = 0.f;
#pragma unroll 4
    for (int ff = 0; ff < 128; ++ff) {
      const int f = (ff + tid * 4) & 127;
      float v = ai[f] + bk[f];
      v = (v >= 0.f) ? v : 0.2f * v;               // leaky_relu, slope 0.2
      acc += v * W2s[f];
    }
    Ss[bb * 100 + i * 10 + k] = acc + b2[h];
  }
  __syncthreads();

  // gather + write (d_out is poisoned, so row j=9 must be written as 0)
  if (tid < 200) {
    const int bb  = tid / 100;
    const int rem = tid % 100;
    const int j   = rem / 10;
    const int ic  = rem % 10;
    float v = 0.f;
    if (j < 9) {
      const int src = j + ((j >= ic) ? 1 : 0);
      v = Ss[bb * 100 + ic * 10 + src];
    }
    out[(size_t)((b0 + bb) * 8 + h) * 100 + j * 10 + ic] = v;
  }
}

// ---------------------------------------------------------------------------
extern "C" void kernel_launch(void* const* d_in, const int* in_sizes, int n_in,
                              void* d_out, int out_size, void* d_ws, size_t ws_size,
                              hipStream_t stream) {
  const float* X  = (const float*)d_in[0]; // object_features (1024,10,2048,1,1)
  // d_in[1]: scene_geometry (unused by the reference math)
  const float* Wp = (const float*)d_in[2];
  const float* bp = (const float*)d_in[3];
  const float* W1 = (const float*)d_in[4];
  const float* b1 = (const float*)d_in[5];
  const float* W2 = (const float*)d_in[6];
  const float* b2 = (const float*)d_in[7];
  // d_in[8]: d_max == 10 (hardcoded in geometry below)

  float* E   = (float*)d_ws;   // 10240 x 128 f32 = 5.24 MB scratch
  float* out = (float*)d_out;  // 1024 x 8 x 10 x 10 f32

  k_emb<<<dim3(160), dim3(256), 0, stream>>>(X, Wp, bp, E);
  k_attn<<<dim3(512, 8), dim3(256), 0, stream>>>(E, W1, b1, W2, b2, out);
}